// DeformableQuantizer_20641612824631
// MI455X (gfx1250) — compile-verified
//
#include <hip/hip_runtime.h>

// DeformableQuantizer for MI455X (gfx1250, wave32).
// Memory-bound target: 201 MB @ 23.3 TB/s ~= 8.6 us. Cross-term einsum
// bsgd,kd->bsgk mapped onto V_WMMA_F32_16X16X4_F32 (K=4 holds group_dim=3+pad)
// to keep VALU issue below the HBM floor. Softmax shift-invariance removes the
// z^2 term; exp via v_exp_f32 (trans pipe), normalization via v_rcp_f32.
// Loop control is forced wave-uniform (readfirstlane + counted loop) so EXEC
// stays all-ones at every WMMA, per ISA 7.12 restrictions.

typedef __attribute__((ext_vector_type(2))) float v2f;
typedef __attribute__((ext_vector_type(8))) float v8f;

#define TEMP   0.3f
#define LOG2E  1.4426950408889634f

// lane gather via LDS-permute hardware (no LDS allocation, wave-local)
__device__ __forceinline__ float laneGather(float v, int srcLane) {
  return __int_as_float(__builtin_amdgcn_ds_bpermute(srcLane << 2, __float_as_int(v)));
}

__global__ __launch_bounds__(256)
void DeformableQuantizer_20641612824631_kernel(
    const float* __restrict__ x,        // [B*S*D] fp32
    const float* __restrict__ delta,    // [8*3]  fp32
    const float* __restrict__ dscale_p, // [1]    fp32
    float* __restrict__ out,            // [B*S*D] fp32
    int nTiles,                         // groups / 32
    long nGroups)
{
  const int  tid     = threadIdx.x;
  const int  lane    = tid & 31;
  // Wave-uniform scalars (SGPR): uniform loop control -> EXEC stays all-ones.
  const int  waveId  = __builtin_amdgcn_readfirstlane(
                          (int)((blockIdx.x * blockDim.x + tid) >> 5));
  const int  nWaves  = __builtin_amdgcn_readfirstlane(
                          (int)((gridDim.x * blockDim.x) >> 5));
  const bool lowHalf = lane < 16;
  const int  nsel    = lane & 15;

  const float ds  = dscale_p[0];
  const float k2t = LOG2E / TEMP;          // 1/(T*ln2)
  const float scale2 = 2.0f * k2t;

  // ---- uniform deformed codebook + per-code bias (scalar/SGPR resident) ----
  // BASE_CODEBOOK row k = (sign(k&4), sign(k&2), sign(k&1))
  float cb[8][3];
  float biasK[8];
#pragma unroll
  for (int k = 0; k < 8; ++k) {
    const float s0 = (k & 4) ? 1.f : -1.f;
    const float s1 = (k & 2) ? 1.f : -1.f;
    const float s2 = (k & 1) ? 1.f : -1.f;
    cb[k][0] = __builtin_fmaf(ds, delta[3 * k + 0], s0);
    cb[k][1] = __builtin_fmaf(ds, delta[3 * k + 1], s1);
    cb[k][2] = __builtin_fmaf(ds, delta[3 * k + 2], s2);
    const float n2 = cb[k][0] * cb[k][0] + cb[k][1] * cb[k][1] + cb[k][2] * cb[k][2];
    biasK[k] = n2 * k2t;                   // ||cb_k||^2 / (T*ln2)
  }

  // ---- constant A operands: cb^T padded into 16x4 (f32 A-layout, 2 VGPRs) ----
  // A layout (ISA 7.12.2, 32-bit A 16x4): VGPR0 = {K0 lanes0-15 | K2 lanes16-31},
  //                                       VGPR1 = {K1 lanes0-15 | K3 lanes16-31}; row M = lane&15.
  // A1: codes in rows 0-7  -> D rows 0-7  land in lanes 0-15  (groups 0-15 of tile)
  // A2: codes in rows 8-15 -> D rows 8-15 land in lanes 16-31 (groups 16-31 of tile)
  float a1_0, a1_1, a2_0, a2_1;
  {
    const int kk = nsel & 7;
    const float s0 = (kk & 4) ? 1.f : -1.f;
    const float s1 = (kk & 2) ? 1.f : -1.f;
    const float s2 = (kk & 1) ? 1.f : -1.f;
    const float c0 = __builtin_fmaf(ds, delta[3 * kk + 0], s0);
    const float c1 = __builtin_fmaf(ds, delta[3 * kk + 1], s1);
    const float c2 = __builtin_fmaf(ds, delta[3 * kk + 2], s2);
    const float e0 = lowHalf ? c0 : c2;    // K=0 (low half) / K=2 (high half)
    const float e1 = lowHalf ? c1 : 0.f;   // K=1 / K=3 pad
    const bool v1 = (nsel < 8), v2 = (nsel >= 8);
    a1_0 = v1 ? e0 : 0.f;  a1_1 = v1 ? e1 : 0.f;
    a2_0 = v2 ? e0 : 0.f;  a2_1 = v2 ? e1 : 0.f;
  }
  const v2f A1 = { a1_0, a1_1 };
  const v2f A2 = { a2_0, a2_1 };

  // ---- one tile = 32 groups = 96 contiguous floats per wave ----
  auto doTile = [&](long tile) {
    const long gbase = tile * 32;                       // first group id
    const float* __restrict__ gx = x + gbase * 3 + (long)lane * 3;
    const float z0 = gx[0], z1 = gx[1], z2 = gx[2];     // coalesced 384B/wave
    __builtin_prefetch(gx + (long)nWaves * 96, 0, 0);   // global_prefetch_b8, next tile

    // B operand (4x16 f32, K-striped like A): group = N column.
    // WMMA1 covers groups gbase+0..15:  low lanes own z0/z1, high lanes need z2 of lane-16.
    const float b1_0 = lowHalf ? z0 : laneGather(z2, nsel);
    const float b1_1 = lowHalf ? z1 : 0.f;
    // WMMA2 covers groups gbase+16..31: low lanes need z0/z1 of lane+16, high lanes own z2.
    const float b2_0 = lowHalf ? laneGather(z0, nsel + 16) : z2;
    const float b2_1 = lowHalf ? laneGather(z1, nsel + 16) : 0.f;
    const v2f B1 = { b1_0, b1_1 };
    const v2f B2 = { b2_0, b2_1 };

    v8f acc0 = {};
    v8f acc1 = {};
    // D = cb(16x4) x groups^T(4x16): D[M=code][N=group]
    acc0 = __builtin_amdgcn_wmma_f32_16x16x4_f32(false, A1, false, B1,
                                                 (short)0, acc0, false, false);
    acc1 = __builtin_amdgcn_wmma_f32_16x16x4_f32(false, A2, false, B2,
                                                 (short)0, acc1, false, false);

    // Each lane now owns all 8 cross terms of exactly its group (merge = cndmask).
    float l[8];
#pragma unroll
    for (int k = 0; k < 8; ++k) {
      const float cr = lowHalf ? acc0[k] : acc1[k];
      l[k] = __builtin_fmaf(cr, scale2, -biasK[k]);     // logits in log2 domain
    }
    float mx = l[0];
#pragma unroll
    for (int k = 1; k < 8; ++k) mx = fmaxf(mx, l[k]);
    float e[8], s = 0.f;
#pragma unroll
    for (int k = 0; k < 8; ++k) {
      e[k] = __builtin_amdgcn_exp2f(l[k] - mx);         // v_exp_f32 (trans pipe)
      s += e[k];
    }
    const float r = __builtin_amdgcn_rcpf(s);
    float q0 = 0.f, q1 = 0.f, q2 = 0.f;
#pragma unroll
    for (int k = 0; k < 8; ++k) {
      q0 = __builtin_fmaf(e[k], cb[k][0], q0);
      q1 = __builtin_fmaf(e[k], cb[k][1], q1);
      q2 = __builtin_fmaf(e[k], cb[k][2], q2);
    }
    float* __restrict__ go = out + gbase * 3 + (long)lane * 3;
    go[0] = q0 * r;
    go[1] = q1 * r;
    go[2] = q2 * r;                                     // STE forward == quantized
  };

  // Uniform counted main loop (SALU branch, no EXEC juggling around WMMA).
  const int nIter = nTiles / nWaves;
#pragma unroll 2
  for (int i = 0; i < nIter; ++i) {
    doTile((long)i * nWaves + waveId);
  }
  // Remainder tiles (uniform predicate per wave).
  {
    const long tile = (long)nIter * nWaves + waveId;
    if (tile < (long)nTiles) doTile(tile);
  }

  // ---- ragged tail groups (0 for the reference shapes; scalar, no WMMA) ----
  const long doneGroups = (long)nTiles * 32;
  if (waveId == 0 && doneGroups + lane < nGroups) {
    const long g = doneGroups + lane;
    const float z0 = x[g * 3 + 0], z1 = x[g * 3 + 1], z2 = x[g * 3 + 2];
    float l[8];
#pragma unroll
    for (int k = 0; k < 8; ++k) {
      float cr = z0 * cb[k][0] + z1 * cb[k][1] + z2 * cb[k][2];
      l[k] = __builtin_fmaf(cr, scale2, -biasK[k]);
    }
    float mx = l[0];
#pragma unroll
    for (int k = 1; k < 8; ++k) mx = fmaxf(mx, l[k]);
    float e[8], s = 0.f;
#pragma unroll
    for (int k = 0; k < 8; ++k) { e[k] = __builtin_amdgcn_exp2f(l[k] - mx); s += e[k]; }
    const float r = __builtin_amdgcn_rcpf(s);
    float q0 = 0.f, q1 = 0.f, q2 = 0.f;
#pragma unroll
    for (int k = 0; k < 8; ++k) {
      q0 = __builtin_fmaf(e[k], cb[k][0], q0);
      q1 = __builtin_fmaf(e[k], cb[k][1], q1);
      q2 = __builtin_fmaf(e[k], cb[k][2], q2);
    }
    out[g * 3 + 0] = q0 * r;
    out[g * 3 + 1] = q1 * r;
    out[g * 3 + 2] = q2 * r;
  }
}

extern "C" void kernel_launch(void* const* d_in, const int* in_sizes, int n_in,
                              void* d_out, int out_size, void* d_ws, size_t ws_size,
                              hipStream_t stream) {
  const float* x      = (const float*)d_in[0];   // [8,4096,768] fp32
  const float* delta  = (const float*)d_in[1];   // [8,3] fp32
  const float* dscale = (const float*)d_in[2];   // scalar fp32
  float* out = (float*)d_out;

  const long nElems  = (long)in_sizes[0];
  const long nGroups = nElems / 3;               // 8,388,608
  const int  nTiles  = (int)(nGroups / 32);      // 262,144 tiles of 32 groups

  // 2048 blocks x 8 wave32 = 16384 waves; 262144/16384 = 16 uniform iterations.
  long maxBlocks = (nTiles + 7) / 8;
  if (maxBlocks < 1) maxBlocks = 1;
  int blocks = 2048;
  if ((long)blocks > maxBlocks) blocks = (int)maxBlocks;

  DeformableQuantizer_20641612824631_kernel<<<blocks, 256, 0, stream>>>(
      x, delta, dscale, out, nTiles, nGroups);
}